// HilbertSimulator_12275016532163
// MI455X (gfx1250) — compile-verified
//
#include <hip/hip_runtime.h>

typedef float v2f __attribute__((ext_vector_type(2)));
typedef float v8f __attribute__((ext_vector_type(8)));

// Padded Z-readout matrix element: Zpad[r][n] = (n<4) ? 1-2*bit_{3-n}(r) : 0
__device__ __forceinline__ float zpad(int r, int n) {
  return (n < 4) ? (1.0f - 2.0f * (float)((r >> (3 - n)) & 1)) : 0.0f;
}

__launch_bounds__(256, 4)
__global__ void qsim_wmma_kernel(const float* __restrict__ x,
                                 const float* __restrict__ w,
                                 float* __restrict__ out,
                                 int nTiles) {
  const int lane = threadIdx.x & 31;
  const int wavesPerBlock = blockDim.x >> 5;
  const int wave = blockIdx.x * wavesPerBlock + (threadIdx.x >> 5);
  const int nWaves = gridDim.x * wavesPerBlock;

  const int n  = lane & 15;         // WMMA N index / batch row within tile
  const int hi = lane >> 4;         // half-wave: selects K block of A/B frags
  const bool upper = (hi != 0);

  // Uniform weight loads -> scalar loads
  const float w0 = w[0], w1 = w[1], w2 = w[2], w3 = w[3];

  // B fragments (loop-invariant): block kb covers K rows 4kb..4kb+3;
  // VGPR0 -> row 4kb+2*hi, VGPR1 -> row 4kb+2*hi+1.
  const int rb = 2 * hi;
  const float b0x = zpad(0 + rb, n),  b0y = zpad(1 + rb, n);
  const float b1x = zpad(4 + rb, n),  b1y = zpad(5 + rb, n);
  const float b2x = zpad(8 + rb, n),  b2y = zpad(9 + rb, n);
  const float b3x = zpad(12 + rb, n), b3y = zpad(13 + rb, n);

  for (int tile = wave; tile < nTiles; tile += nWaves) {
    const int m = tile * 16 + n;  // batch element this lane evaluates
    const float4 xv = reinterpret_cast<const float4*>(x)[m];

    // RY(x_i) then RY(w_i) on qubit i == RY(x_i + w_i); product state.
    const float t0 = xv.x + w0, t1 = xv.y + w1, t2 = xv.z + w2, t3 = xv.w + w3;
    const float c0 = __cosf(t0), s0 = __sinf(t0);
    const float c1 = __cosf(t1), s1 = __sinf(t1);
    const float c2 = __cosf(t2), s2 = __sinf(t2);
    const float c3 = __cosf(t3), s3 = __sinf(t3);

    // Squared pair factors: prob[m] = e01sq[m>>2] * e23sq[m&3]
    const float g0 = c0 * c1, g1 = c0 * s1, g2 = s0 * c1, g3 = s0 * s1;
    const float h0 = c2 * c3, h1 = c2 * s3, h2 = s2 * c3, h3 = s2 * s3;
    const float G0 = g0 * g0, G1 = g1 * g1, G2 = g2 * g2, G3 = g3 * g3;
    const float H0 = h0 * h0, H1 = h1 * h1, H2 = h2 * h2, H3 = h3 * h3;

    // probs after CNOT chain, permutation Q applied BY HAND (no arrays):
    // Q = [0,13,3,14,6,11,5,8,12,1,15,2,10,7,9,4];  p_k = G[Q>>2]*H[Q&3]
    const float p0  = G0 * H0;  // Q=0
    const float p1  = G3 * H1;  // Q=13
    const float p2  = G0 * H3;  // Q=3
    const float p3  = G3 * H2;  // Q=14
    const float p4  = G1 * H2;  // Q=6
    const float p5  = G2 * H3;  // Q=11
    const float p6  = G1 * H1;  // Q=5
    const float p7  = G2 * H0;  // Q=8
    const float p8  = G3 * H0;  // Q=12
    const float p9  = G0 * H1;  // Q=1
    const float p10 = G3 * H3;  // Q=15
    const float p11 = G0 * H2;  // Q=2
    const float p12 = G2 * H2;  // Q=10
    const float p13 = G1 * H3;  // Q=7
    const float p14 = G2 * H1;  // Q=9
    const float p15 = G1 * H0;  // Q=4

    // A fragments: lanes<16 carry K = {4kb, 4kb+1}; lanes>=16 K = {4kb+2, 4kb+3}
    v2f a0, a1, a2, a3, b;
    a0.x = upper ? p2  : p0;   a0.y = upper ? p3  : p1;
    a1.x = upper ? p6  : p4;   a1.y = upper ? p7  : p5;
    a2.x = upper ? p10 : p8;   a2.y = upper ? p11 : p9;
    a3.x = upper ? p14 : p12;  a3.y = upper ? p15 : p13;

    // out_tile(16x4) = P(16x16) x Zpad(16x16) via 4 chained f32 WMMAs (K=4 ea)
    v8f acc = {};
    b.x = b0x; b.y = b0y;
    acc = __builtin_amdgcn_wmma_f32_16x16x4_f32(false, a0, false, b, (short)0,
                                                acc, false, false);
    b.x = b1x; b.y = b1y;
    acc = __builtin_amdgcn_wmma_f32_16x16x4_f32(false, a1, false, b, (short)0,
                                                acc, false, false);
    b.x = b2x; b.y = b2y;
    acc = __builtin_amdgcn_wmma_f32_16x16x4_f32(false, a2, false, b, (short)0,
                                                acc, false, false);
    b.x = b3x; b.y = b3y;
    acc = __builtin_amdgcn_wmma_f32_16x16x4_f32(false, a3, false, b, (short)0,
                                                acc, false, false);

    // D layout: lane holds N = n, rows M = v + 8*hi in acc[v].
    // Columns n<4 are real output; a tile's 64 floats are one contiguous
    // 256-byte block of `out`, so these masked stores coalesce in L2.
    if (n < 4) {
      float* o = out + (size_t)tile * 64 + n + (size_t)hi * 32;
      o[0]  = acc[0];
      o[4]  = acc[1];
      o[8]  = acc[2];
      o[12] = acc[3];
      o[16] = acc[4];
      o[20] = acc[5];
      o[24] = acc[6];
      o[28] = acc[7];
    }
  }
}

extern "C" void kernel_launch(void* const* d_in, const int* in_sizes, int n_in,
                              void* d_out, int out_size, void* d_ws, size_t ws_size,
                              hipStream_t stream) {
  const float* x = (const float*)d_in[0];   // (BATCH, 4) float32
  const float* w = (const float*)d_in[1];   // (1, 4) float32
  float* out = (float*)d_out;               // (BATCH, 4) float32

  const int batch  = in_sizes[0] / 4;
  const int nTiles = batch / 16;            // BATCH = 2^21 -> multiple of 16

  const int threads = 256;                  // 8 wave32s per block
  const int wavesPerBlock = threads / 32;
  int blocks = 1024;                        // 8192 waves, ~16 tiles each
  const int maxBlocks = (nTiles + wavesPerBlock - 1) / wavesPerBlock;
  if (blocks > maxBlocks) blocks = maxBlocks;
  if (blocks < 1) blocks = 1;

  qsim_wmma_kernel<<<blocks, threads, 0, stream>>>(x, w, out, nTiles);
}